// Detect_31568009625973
// MI455X (gfx1250) — compile-verified
//
#include <hip/hip_runtime.h>
#include <hip/hip_bf16.h>

typedef __attribute__((ext_vector_type(2))) float v2f;
typedef __attribute__((ext_vector_type(8))) float v8f;

#define KCHUNK 64
#define PTILE  32
#define NOUT   255     // 3 anchors * 85
#define NO     85
#define BS_TOTAL_ROWS 25200

// One workgroup: 16 waves (512 thr), one 32-pixel tile, all 256 padded rows.
// Wave w owns M rows [16w,16w+16) and computes two 16x16 N-subtiles.
//
// LDS layout (pair-swizzled so one b64 load == one B fragment):
//   element (kk, p) lives at float offset (kk/2)*64 + p*2 + (kk&1)
//   -> fragment {K=kk, K=kk+1} for column p is contiguous 8 bytes.
__global__ __launch_bounds__(512) void detect_level_kernel(
    const float* __restrict__ x,     // [16, C, ny, nx]
    const float* __restrict__ w,     // [255, C]
    const float* __restrict__ bias,  // [255]
    float* __restrict__ out,         // [16, 25200, 85]
    int C, int nynx, int nx, float stride_px,
    float aw0, float ah0, float aw1, float ah1, float aw2, float ah2,
    int rowbase)
{
    __shared__ float lds[KCHUNK * PTILE];   // 8KB

    const int tid    = threadIdx.x;
    const int lane   = tid & 31;
    const int waveid = tid >> 5;            // 0..15 == M tile index
    const int p0     = blockIdx.x * PTILE;  // pixel tile base

    const int col   = lane & 15;            // N within subtile / M row for A
    const int khalf = lane >> 4;            // 0: K={0,1}, 1: K={2,3}

    // A operand: clamp padded row 255 -> 254. Its products land only in
    // D-row 255 which is never stored, so no masking is required.
    const int orow = waveid * 16 + col;
    const int orow_c = (orow < NOUT) ? orow : (NOUT - 1);
    const float* wrow = w + (size_t)orow_c * (size_t)C;

    // Cooperative fill: thread t handles K-pair fk2 = t/16 (rows 2*fk2, 2*fk2+1)
    // and pixels fpp, fpp+1 (fpp = (t%16)*2). Two float2 global loads
    // (half-wave covers 32 contiguous pixels = 128B) + one b128 LDS store.
    const int fk2 = tid >> 4;               // 0..31
    const int fpp = (tid & 15) * 2;         // 0..30
    const int fpg = p0 + fpp;
    const int fbb = fpg / nynx;
    const int frr = fpg - fbb * nynx;
    const float* fsrc = x + ((size_t)fbb * (size_t)C + (size_t)(2 * fk2)) * (size_t)nynx + frr;
    float* fdst = &lds[fk2 * 64 + fpp * 2];

    // B fragment base for this lane.
    const float* bbase = &lds[col * 2];

    v8f acc0 = {};
    v8f acc1 = {};

    for (int k0 = 0; k0 < C; k0 += KCHUNK) {
        // Stage chunk (2x2 register transpose into pair-swizzled LDS).
        v2f g0 = *(const v2f*)(fsrc + (size_t)k0 * (size_t)nynx);
        v2f g1 = *(const v2f*)(fsrc + (size_t)(k0 + 1) * (size_t)nynx);
        __syncthreads();                    // prior chunk's LDS reads done
        float4 stg;
        stg.x = g0.x; stg.y = g1.x; stg.z = g0.y; stg.w = g1.y;
        *(float4*)fdst = stg;
        __syncthreads();                    // chunk visible to all waves

        // Preload all A fragments for this chunk (16 x b64, unconditional).
        v2f a[KCHUNK / 4];
        #pragma unroll
        for (int i = 0; i < KCHUNK / 4; ++i) {
            a[i] = *(const v2f*)(wrow + k0 + 4 * i + 2 * khalf);
        }

        // 32 WMMAs per chunk; each B fragment is a single b64 LDS load.
        #pragma unroll
        for (int i = 0; i < KCHUNK / 4; ++i) {
            const float* bp = bbase + (2 * i + khalf) * 64;
            v2f b0 = *(const v2f*)(bp);
            v2f b1 = *(const v2f*)(bp + 32);
            acc0 = __builtin_amdgcn_wmma_f32_16x16x4_f32(
                false, a[i], false, b0, (short)0, acc0, false, false);
            acc1 = __builtin_amdgcn_wmma_f32_16x16x4_f32(
                false, a[i], false, b1, (short)0, acc1, false, false);
        }
    }

    // --- Epilogue: bias + sigmoid + YOLO decode + store ---
    // C/D layout: VGPR r -> M = mbase + r, N = col.
    const int mbase = waveid * 16 + khalf * 8;

    #pragma unroll
    for (int n = 0; n < 2; ++n) {
        const v8f acc = n ? acc1 : acc0;
        const int pix = p0 + n * 16 + col;
        const int b   = pix / nynx;
        const int rem = pix - b * nynx;
        const int gy  = rem / nx;
        const int gx  = rem - gy * nx;

        #pragma unroll
        for (int r = 0; r < 8; ++r) {
            const int o = mbase + r;        // output channel 0..255
            if (o < NOUT) {
                float h = acc[r] + bias[o];
                float y = 1.0f / (1.0f + __expf(-h));
                const int a_idx = o / NO;
                const int oo    = o - a_idx * NO;
                const float anw = (a_idx == 0) ? aw0 : ((a_idx == 1) ? aw1 : aw2);
                const float anh = (a_idx == 0) ? ah0 : ((a_idx == 1) ? ah1 : ah2);
                float val;
                if (oo == 0)      val = (y * 2.0f - 0.5f + (float)gx) * stride_px;
                else if (oo == 1) val = (y * 2.0f - 0.5f + (float)gy) * stride_px;
                else if (oo == 2) { float t = y * 2.0f; val = t * t * anw; }
                else if (oo == 3) { float t = y * 2.0f; val = t * t * anh; }
                else              val = y;
                const size_t row = (size_t)b * BS_TOTAL_ROWS + (size_t)rowbase
                                 + (size_t)a_idx * (size_t)nynx + (size_t)rem;
                out[row * NO + oo] = val;
            }
        }
    }
}

extern "C" void kernel_launch(void* const* d_in, const int* in_sizes, int n_in,
                              void* d_out, int out_size, void* d_ws, size_t ws_size,
                              hipStream_t stream) {
    const float* x0 = (const float*)d_in[0];
    const float* x1 = (const float*)d_in[1];
    const float* x2 = (const float*)d_in[2];
    const float* w0 = (const float*)d_in[3];
    const float* w1 = (const float*)d_in[4];
    const float* w2 = (const float*)d_in[5];
    const float* b0 = (const float*)d_in[6];
    const float* b1 = (const float*)d_in[7];
    const float* b2 = (const float*)d_in[8];
    float* out = (float*)d_out;

    dim3 block(512);
    // blocks = (16 * nynx) / 32
    detect_level_kernel<<<3200, block, 0, stream>>>(
        x0, w0, b0, out, 256, 6400, 80, 8.0f,
        10.0f, 13.0f, 16.0f, 30.0f, 33.0f, 23.0f, 0);
    detect_level_kernel<<<800, block, 0, stream>>>(
        x1, w1, b1, out, 512, 1600, 40, 16.0f,
        30.0f, 61.0f, 62.0f, 45.0f, 59.0f, 119.0f, 19200);
    detect_level_kernel<<<200, block, 0, stream>>>(
        x2, w2, b2, out, 1024, 400, 20, 32.0f,
        116.0f, 90.0f, 156.0f, 198.0f, 373.0f, 326.0f, 24000);
}